// DirectVoxGO_13494787244175
// MI455X (gfx1250) — compile-verified
//
#include <hip/hip_runtime.h>
#include <hip/hip_bf16.h>
#include <stdint.h>

#define NR 16384
#define NS 256
#define RES 160
#define CH ((size_t)RES * RES * RES) /* 4,096,000 floats per k0 channel */

#define WAVES_PER_BLOCK 8
#define BLOCK (WAVES_PER_BLOCK * 32)

/* log(1/(1-1e-6) - 1) computed in double precision */
#define ACT_SHIFT (-13.815509557964274f)

#if __has_builtin(__builtin_amdgcn_tensor_load_to_lds) && \
    __has_builtin(__builtin_amdgcn_s_wait_tensorcnt) &&   \
    __has_builtin(__builtin_amdgcn_readfirstlane)
#define USE_TDM 1
#else
#define USE_TDM 0
#endif

typedef unsigned int u32x4 __attribute__((ext_vector_type(4)));
typedef int i32x4 __attribute__((ext_vector_type(4)));
typedef int i32x8 __attribute__((ext_vector_type(8)));

__device__ __forceinline__ float sigmoidf_fast(float x) {
    return 1.0f / (1.0f + __expf(-x));
}

/* 8-corner trilinear lerp; base index b = ((x0*RES)+y0)*RES + z0, z contiguous */
__device__ __forceinline__ float tri8(const float* __restrict__ g, size_t b,
                                      float fx, float fy, float fz) {
    const int OY = RES;
    const int OX = RES * RES;
    float c000 = g[b];
    float c001 = g[b + 1];
    float c010 = g[b + OY];
    float c011 = g[b + OY + 1];
    float c100 = g[b + OX];
    float c101 = g[b + OX + 1];
    float c110 = g[b + OX + OY];
    float c111 = g[b + OX + OY + 1];
    float c00 = fmaf(fz, c001 - c000, c000);
    float c01 = fmaf(fz, c011 - c010, c010);
    float c10 = fmaf(fz, c101 - c100, c100);
    float c11 = fmaf(fz, c111 - c110, c110);
    float c0 = fmaf(fy, c01 - c00, c00);
    float c1 = fmaf(fy, c11 - c10, c10);
    return fmaf(fx, c1 - c0, c0);
}

__global__ __launch_bounds__(BLOCK) void dvgo_render(
    const float* __restrict__ pts,   /* [NR][NS][3] */
    const float* __restrict__ dens,  /* [RES][RES][RES] */
    const float* __restrict__ k0,    /* [3][RES][RES][RES] */
    float* __restrict__ out)         /* [NR][3] */
{
    const int lane = threadIdx.x & 31;
    const int w = threadIdx.x >> 5;
    const int ray = blockIdx.x * WAVES_PER_BLOCK + w;

#if USE_TDM
    __shared__ float spts[WAVES_PER_BLOCK][NS * 3];
    {
        /* Wave-uniform values for the SGPR-resident tensor descriptor */
        int uray = __builtin_amdgcn_readfirstlane(ray);
        __attribute__((address_space(3))) float* lp =
            (__attribute__((address_space(3))) float*)&spts[w][0];
        unsigned ulds = (unsigned)(unsigned long long)lp;
        ulds = (unsigned)__builtin_amdgcn_readfirstlane((int)ulds);
        unsigned long long ga = (unsigned long long)(uintptr_t)pts +
                                (unsigned long long)uray * (unsigned long long)(NS * 3 * 4);

        /* D# group 0: count=1 | lds_addr | global_addr(57b) | type=2 */
        u32x4 g0;
        g0[0] = 1u;
        g0[1] = ulds;
        g0[2] = (unsigned)(ga & 0xFFFFFFFFu);
        g0[3] = (unsigned)((ga >> 32) & 0x1FFFFFFu) | (2u << 30);

        /* D# group 1: 1-row tile of 768 dwords (one ray's points, 3 KB) */
        i32x8 g1;
        g1[0] = 2 << 16;                       /* workgroup_mask=0, data_size=4B */
        g1[1] = (int)(((unsigned)(NS * 3)) << 16);  /* tensor_dim0 lo16 */
        g1[2] = (int)(1u << 16);               /* tensor_dim0 hi=0 | tensor_dim1=1 */
        g1[3] = (int)(((unsigned)(NS * 3)) << 16);  /* tile_dim0 = 768 */
        g1[4] = 1;                             /* tile_dim1=1, tile_dim2=0 */
        g1[5] = NS * 3;                        /* tensor_dim0_stride lo32 */
        g1[6] = 0;
        g1[7] = 0;
        i32x4 g2 = {0, 0, 0, 0};
        i32x4 g3 = {0, 0, 0, 0};
#if defined(__clang_major__) && (__clang_major__ >= 23)
        i32x8 g4 = {0, 0, 0, 0, 0, 0, 0, 0};
        __builtin_amdgcn_tensor_load_to_lds(g0, g1, g2, g3, g4, 0);
#else
        __builtin_amdgcn_tensor_load_to_lds(g0, g1, g2, g3, 0);
#endif
        __builtin_amdgcn_s_wait_tensorcnt(0);
    }
#else
    const float* __restrict__ rayp = pts + (size_t)ray * (NS * 3);
#endif

    float T = 1.0f;               /* running transmittance (cumprod of clipped 1-alpha) */
    float accR = 0.0f, accG = 0.0f, accB = 0.0f;

    for (int base = 0; base < NS; base += 32) {
        const int s = base + lane;
#if USE_TDM
        float px = spts[w][s * 3 + 0];
        float py = spts[w][s * 3 + 1];
        float pz = spts[w][s * 3 + 2];
#else
        float px = rayp[s * 3 + 0];
        float py = rayp[s * 3 + 1];
        float pz = rayp[s * 3 + 2];
#endif
        /* continuous voxel index: (p - xyz_min)/(max-min) * (RES-1) = (p+1)*79.5 */
        float ix = fminf(fmaxf((px + 1.0f) * 79.5f, 0.0f), (float)(RES - 1));
        float iy = fminf(fmaxf((py + 1.0f) * 79.5f, 0.0f), (float)(RES - 1));
        float iz = fminf(fmaxf((pz + 1.0f) * 79.5f, 0.0f), (float)(RES - 1));
        float fx0 = fminf(floorf(ix), (float)(RES - 2));
        float fy0 = fminf(floorf(iy), (float)(RES - 2));
        float fz0 = fminf(floorf(iz), (float)(RES - 2));
        int x0 = (int)fx0, y0 = (int)fy0, z0 = (int)fz0;
        float fx = ix - fx0, fy = iy - fy0, fz = iz - fz0;
        size_t b = ((size_t)x0 * RES + (size_t)y0) * RES + (size_t)z0;

        float d = tri8(dens, b, fx, fy, fz);
        float r = sigmoidf_fast(tri8(k0, b, fx, fy, fz));
        float g = sigmoidf_fast(tri8(k0 + CH, b, fx, fy, fz));
        float bb = sigmoidf_fast(tri8(k0 + 2 * CH, b, fx, fy, fz));

        /* 1 - exp(-softplus(x)) == sigmoid(x); 1-alpha == sigmoid(-x) */
        float xact = d + ACT_SHIFT;
        float alpha = sigmoidf_fast(xact);
        float om = fmaxf(1.0f / (1.0f + __expf(xact)), 1e-10f);

        /* wave32 inclusive product scan (Kogge-Stone) over clipped (1-alpha) */
        float scan = om;
#pragma unroll
        for (int off = 1; off < 32; off <<= 1) {
            float v = __shfl_up(scan, off, 32);
            if (lane >= off) scan *= v;
        }
        float excl = __shfl_up(scan, 1, 32);
        if (lane == 0) excl = 1.0f;

        float wgt = alpha * T * excl;   /* alpha_s * cumprod_{j<s}(1-alpha_j) */
        accR = fmaf(wgt, r, accR);
        accG = fmaf(wgt, g, accG);
        accB = fmaf(wgt, bb, accB);

        T *= __shfl(scan, 31, 32);      /* carry transmittance to next chunk */
    }

    /* wave-wide butterfly sum */
#pragma unroll
    for (int off = 16; off > 0; off >>= 1) {
        accR += __shfl_xor(accR, off, 32);
        accG += __shfl_xor(accG, off, 32);
        accB += __shfl_xor(accB, off, 32);
    }
    if (lane == 0) {
        out[(size_t)ray * 3 + 0] = accR + T;  /* + alphainv_cum[..., -1] */
        out[(size_t)ray * 3 + 1] = accG + T;
        out[(size_t)ray * 3 + 2] = accB + T;
    }
}

extern "C" void kernel_launch(void* const* d_in, const int* in_sizes, int n_in,
                              void* d_out, int out_size, void* d_ws, size_t ws_size,
                              hipStream_t stream) {
    (void)in_sizes; (void)n_in; (void)out_size; (void)d_ws; (void)ws_size;
    const float* pts  = (const float*)d_in[0];
    const float* dens = (const float*)d_in[1];
    const float* k0   = (const float*)d_in[2];
    float* out = (float*)d_out;

    dim3 grid(NR / WAVES_PER_BLOCK);  /* 2048 blocks */
    dim3 block(BLOCK);                /* 256 threads = 8 wave32 */
    hipLaunchKernelGGL(dvgo_render, grid, block, 0, stream, pts, dens, k0, out);
}